// YoloLoss_37778532335632
// MI455X (gfx1250) — compile-verified
//
#include <hip/hip_runtime.h>
#include <hip/hip_bf16.h>

// YOLO loss, MI455X (gfx1250, wave32).
// Stage 1: async global->LDS staging (ASYNCcnt path) + per-batch loss in d_ws.
// Stage 2: single-wave WMMA f32 16x16x4 reduction of the 16384 partials.

#define S_      7
#define BN_     2
#define C_      20
#define G_      8
#define CELLS_  (S_*S_)            // 49
#define SSB_    (CELLS_*BN_)       // 98
#define CSTRIDE (C_ + 5*BN_)       // 30
#define ROW_    (CELLS_*CSTRIDE)   // 1470
#define IMG_INV (1.0f/448.0f)
#define BATCH_  16384
#define WPB     8                  // waves per block
#define BLOCK_  (WPB*32)

typedef __attribute__((ext_vector_type(2))) float v2f;
typedef __attribute__((ext_vector_type(8))) float v8f;

__device__ __forceinline__ float wave_sum_f(float v) {
    #pragma unroll
    for (int off = 16; off > 0; off >>= 1) v += __shfl_xor(v, off, 32);
    return v;
}
__device__ __forceinline__ int wave_sum_i(int v) {
    #pragma unroll
    for (int off = 16; off > 0; off >>= 1) v += __shfl_xor(v, off, 32);
    return v;
}

// CDNA5 async global->LDS copy (tracked by ASYNCcnt, LDS written directly,
// no VGPR staging). Rows are only 8B-aligned (5880 B stride) -> use B64.
__device__ __forceinline__ void async_ld_b64(unsigned lds_addr, const float* g) {
    asm volatile("global_load_async_to_lds_b64 %0, %1, off"
                 :: "v"(lds_addr), "v"(g) : "memory");
}
__device__ __forceinline__ void wait_async0() {
    asm volatile("s_wait_asynccnt 0x0" ::: "memory");
}

__global__ __launch_bounds__(BLOCK_) void yolo_stage1(
    const float* __restrict__ pred, const float* __restrict__ labels,
    float* __restrict__ partial)
{
    __shared__ float sPred[WPB][ROW_];
    __shared__ float sLab[WPB][G_*5];

    const int lane = threadIdx.x & 31;
    const int wave = threadIdx.x >> 5;
    const int b = blockIdx.x * WPB + wave;

    const float* prow = pred + (size_t)b * ROW_;
    const float* lrow = labels + (size_t)b * (G_*5);

    // gfx1250 global_prefetch_b8: hint L2 about this row before the copies
    __builtin_prefetch(prow + lane * 46, 0, 3);

    // Stage the whole 5880 B row + 160 B label row straight into LDS.
    const unsigned ldsPred = (unsigned)(size_t)&sPred[wave][0];
    const unsigned ldsLab  = (unsigned)(size_t)&sLab[wave][0];
    #pragma unroll 4
    for (int i = lane * 2; i < ROW_; i += 64)           // 735 float-pairs
        async_ld_b64(ldsPred + (unsigned)i * 4u, prow + i);
    if (lane < (G_*5)/2)                                 // 20 float-pairs
        async_ld_b64(ldsLab + (unsigned)lane * 8u, lrow + lane * 2);
    wait_async0();   // per-wave: each wave only reads its own LDS region

    const float* P = sPred[wave];
    const float* L = sLab[wave];

    // --- per-GT work: lanes 0..7 are the real GTs, 8..31 compute duplicates
    const int g = lane & 7;
    const float lx1 = L[g*5+1], ly1 = L[g*5+2], lx2 = L[g*5+3], ly2 = L[g*5+4];
    const int   lcls = (int)L[g*5+0];
    const float cx = (lx1 + lx2) * 0.5f * IMG_INV;
    const float cy = (ly1 + ly2) * 0.5f * IMG_INV;
    const float gw = fminf(fmaxf((lx2 - lx1) * IMG_INV, 1e-4f), 1.0f);
    const float gh = fminf(fmaxf((ly2 - ly1) * IMG_INV, 1e-4f), 1.0f);
    const int cellx = (int)fminf(fmaxf(cx * (float)S_, 0.0f), (float)(S_-1));
    const int celly = (int)fminf(fmaxf(cy * (float)S_, 0.0f), (float)(S_-1));
    const int cbase = (celly*S_ + cellx) * CSTRIDE;

    const float g_x1 = cx - gw*0.5f, g_x2 = cx + gw*0.5f;
    const float g_y1 = cy - gh*0.5f, g_y2 = cy + gh*0.5f;
    const float ga = fmaxf(g_x2 - g_x1, 0.0f) * fmaxf(g_y2 - g_y1, 0.0f);

    float iou0 = 0.0f, iou1 = 0.0f;
    #pragma unroll
    for (int bx = 0; bx < BN_; ++bx) {
        const float px = fminf(fmaxf(P[cbase + C_ + bx*5 + 0], 0.0f), 1.0f);
        const float py = fminf(fmaxf(P[cbase + C_ + bx*5 + 1], 0.0f), 1.0f);
        const float pw = fminf(fmaxf(P[cbase + C_ + bx*5 + 2], 1e-4f), 1.0f);
        const float ph = fminf(fmaxf(P[cbase + C_ + bx*5 + 3], 1e-4f), 1.0f);
        const float p_x1 = px - pw*0.5f, p_x2 = px + pw*0.5f;
        const float p_y1 = py - ph*0.5f, p_y2 = py + ph*0.5f;
        const float iw = fmaxf(fminf(p_x2, g_x2) - fmaxf(p_x1, g_x1), 0.0f);
        const float ih = fmaxf(fminf(p_y2, g_y2) - fmaxf(p_y1, g_y1), 0.0f);
        const float inter = iw * ih;
        const float pa = fmaxf(p_x2 - p_x1, 0.0f) * fmaxf(p_y2 - p_y1, 0.0f);
        const float uni = fmaxf(pa + ga - inter, 1e-6f);
        const float iou = inter / uni;
        if (bx == 0) iou0 = iou; else iou1 = iou;
    }
    const int best = (iou0 >= iou1) ? 0 : 1;
    const float iou_best = (best == 0) ? iou0 : iou1;
    const int fidx = (celly*S_ + cellx) * BN_ + best;

    // focal-weighted CE over the 20 class logits of this cell
    float mx = -3.0e38f;
    #pragma unroll
    for (int k = 0; k < C_; ++k) mx = fmaxf(mx, P[cbase + k]);
    float se = 0.0f;
    #pragma unroll
    for (int k = 0; k < C_; ++k) se += __expf(P[cbase + k] - mx);
    const float ce = -((P[cbase + lcls] - mx) - __logf(se));
    const float pt = __expf(-ce);
    const float clsl = (1.0f - pt) * (1.0f - pt) * ce;

    float box_cls_part = (lane < G_) ? ((1.0f - iou_best) + clsl) : 0.0f;

    // broadcast the 8 flat obj indices to every lane
    int fall[G_];
    #pragma unroll
    for (int k = 0; k < G_; ++k) fall[k] = __shfl(fidx, k, 32);

    // --- conf sweep over 98 slots
    float obj_sse = 0.0f, noobj_sse = 0.0f;
    int nobj = 0;
    for (int j = lane; j < SSB_; j += 32) {
        const int cell = j >> 1, bx = j & 1;
        const float conf = P[cell*CSTRIDE + C_ + bx*5 + 4];
        bool is_obj = false;
        #pragma unroll
        for (int k = 0; k < G_; ++k) is_obj |= (fall[k] == j);
        if (is_obj) { obj_sse += (conf - 1.0f)*(conf - 1.0f); nobj += 1; }
        else        { noobj_sse += conf * conf; }
    }
    obj_sse   = wave_sum_f(obj_sse);
    noobj_sse = wave_sum_f(noobj_sse);
    nobj      = wave_sum_i(nobj);
    const float bc = wave_sum_f(box_cls_part);

    if (lane == 0) {
        const float obj_mse   = obj_sse   / fmaxf((float)nobj, 1.0f);
        const float noobj_mse = noobj_sse / fmaxf((float)(SSB_ - nobj), 1.0f);
        partial[b] = bc + 5.0f*obj_mse + 0.5f*noobj_mse;
    }
}

// Single-wave final reduction using the f32 matrix engine:
// A = ones(16x4), B = 64 partials per step, C accumulates column sums in f32.
__global__ __launch_bounds__(32) void yolo_stage2(
    const float* __restrict__ partial, float* __restrict__ out)
{
    const int lane = threadIdx.x;
    v2f a; a.x = 1.0f; a.y = 1.0f;
    v8f c = {};
    for (int base = 0; base < BATCH_; base += 64) {
        v2f bv;
        bv.x = partial[base + lane*2 + 0];
        bv.y = partial[base + lane*2 + 1];
        c = __builtin_amdgcn_wmma_f32_16x16x4_f32(
                /*neg_a=*/false, a, /*neg_b=*/false, bv,
                /*c_mod=*/(short)0, c, /*reuse_a=*/false, /*reuse_b=*/false);
    }
    // row M=0 of D lives in c[0] of lanes 0..15 (one column sum per lane)
    float v = (lane < 16) ? c[0] : 0.0f;
    v = wave_sum_f(v);
    if (lane == 0) out[0] = v;
}

extern "C" void kernel_launch(void* const* d_in, const int* in_sizes, int n_in,
                              void* d_out, int out_size, void* d_ws, size_t ws_size,
                              hipStream_t stream) {
    const float* pred   = (const float*)d_in[0];
    const float* labels = (const float*)d_in[1];
    float* partial = (float*)d_ws;                 // 16384 floats = 64 KB
    yolo_stage1<<<BATCH_/WPB, BLOCK_, 0, stream>>>(pred, labels, partial);
    yolo_stage2<<<1, 32, 0, stream>>>(partial, (float*)d_out);
}